// FiLM_1924145349287
// MI455X (gfx1250) — compile-verified
//
#include <hip/hip_runtime.h>
#include <math.h>

// ---------------------------------------------------------------------------
// FiLM reference collapsed to a per-channel linear operator.
//   out[b,t,e] = (G[t,:]·xn[b,e,:] + mlp_b)*std[b,e] + mean[b,e]
// G built on-device from weights with f32 WMMA (v_wmma_f32_16x16x4_f32).
// ---------------------------------------------------------------------------

typedef __attribute__((ext_vector_type(2))) float v2f;
typedef __attribute__((ext_vector_type(8))) float v8f;

#define PI_F 3.14159265358979323846f

__device__ __forceinline__ v8f wmma_f32_4(v2f a, v2f b, v8f c) {
  // D = A(16x4 f32) * B(4x16 f32) + C(16x16 f32)
  return __builtin_amdgcn_wmma_f32_16x16x4_f32(
      /*neg_a=*/false, a, /*neg_b=*/false, b,
      /*c_mod=*/(short)0, c, /*reuse_a=*/false, /*reuse_b=*/false);
}

// force a wave-uniform value into an SGPR so loops get scalar control flow
__device__ __forceinline__ int uniform_i32(int v) {
  return __builtin_amdgcn_readfirstlane(v);
}

// ---------------------------------------------------------------------------
// 1) Per-channel mean/std over L=720 and normalized copy. One wave / channel.
// ---------------------------------------------------------------------------
__global__ __launch_bounds__(256) void norm_kernel(
    const float* __restrict__ x, float* __restrict__ xn,
    float* __restrict__ meanA, float* __restrict__ stdA) {
  int wave = uniform_i32(blockIdx.x * (blockDim.x >> 5) + (threadIdx.x >> 5));
  if (wave >= 1024) return;                   // B*E = 16*64
  int b = wave >> 6, e = wave & 63;
  int lane = threadIdx.x & 31;
  const float* xp = x + (size_t)b * 720 * 64 + e;
  float s1 = 0.f, s2 = 0.f;
  for (int p = lane; p < 720; p += 32) {
    float v = xp[(size_t)p * 64];
    s1 += v; s2 += v * v;
  }
  for (int off = 16; off > 0; off >>= 1) {
    s1 += __shfl_down(s1, off, 32);
    s2 += __shfl_down(s2, off, 32);
  }
  float mean = __shfl(s1, 0, 32) * (1.0f / 720.0f);
  float var  = __shfl(s2, 0, 32) * (1.0f / 720.0f) - mean * mean;
  float stdv = sqrtf(var + 1e-5f);
  if (lane == 0) { meanA[wave] = mean; stdA[wave] = stdv; }
  float inv = 1.0f / stdv;
  float* xo = xn + (size_t)b * 720 * 64 + e;
  for (int p = lane; p < 720; p += 32)
    xo[(size_t)p * 64] = (xp[(size_t)p * 64] - mean) * inv;
}

// ---------------------------------------------------------------------------
// 2) Sequential chunk of impulse responses P[k]=Ad^k Bd, k=0..63.
//    One workgroup; Ad staged in 256KB dynamic LDS (fits 320KB/WGP).
// ---------------------------------------------------------------------------
__global__ __launch_bounds__(256) void seqP_kernel(
    const float* __restrict__ Ad, const float* __restrict__ Bd,
    float* __restrict__ Pmat, int nrows) {
  extern __shared__ float smem[];             // 65536 (Ad) + 256 (prev)
  float* sAd = smem;
  float* prev = smem + 65536;
  int tid = threadIdx.x;                      // 256 threads
  for (int idx = tid; idx < 65536; idx += 256) sAd[idx] = Ad[idx];
  float p0 = Bd[tid];
  prev[tid] = p0;
  Pmat[tid] = p0;
  __syncthreads();
  for (int k = 1; k < nrows; ++k) {
    float s = 0.f;
    // stagger j to avoid 32-way LDS bank conflicts (row stride 256 == 0 mod 64)
    for (int j = 0; j < 256; ++j) {
      int jj = (j + tid) & 255;
      s += sAd[tid * 256 + jj] * prev[jj];
    }
    Pmat[k * 256 + tid] = s;
    __syncthreads();
    prev[tid] = s;
    __syncthreads();
  }
}

// ---------------------------------------------------------------------------
// 3) f32 WMMA GEMM, transB specialized at compile time.
//    One wave computes a 16x32 C tile (two accumulator chains share the A
//    fragment -> 2x arithmetic intensity + WMMA ILP). Branch-free inner loop,
//    pointer-bump addressing. M mult of 16, N mult of 32, K mult of 4.
// ---------------------------------------------------------------------------
template <int TRANSB>
__global__ __launch_bounds__(256) void gemm_wmma_f32(
    const float* __restrict__ A, int lda,
    const float* __restrict__ B, int ldb,
    float* __restrict__ C, int ldc,
    int N, int K, int total_tiles) {
  int tile = uniform_i32(blockIdx.x * (blockDim.x >> 5) + (threadIdx.x >> 5));
  if (tile >= total_tiles) return;
  int ntn = N >> 5;                           // N-tiles of 32
  int tm = tile / ntn, tn = tile - tm * ntn;
  int lane = threadIdx.x & 31;
  int row  = tm * 16 + (lane & 15);           // A fragment row (M)
  int col0 = tn * 32 + (lane & 15);           // B fragment cols (N)
  int col1 = col0 + 16;
  int khalf = (lane >> 4) * 2;                // lanes 16-31 hold K=2,3

  const float* ap = A + (size_t)row * lda + khalf;
  const float* bp0;
  const float* bp1;
  if (TRANSB) {
    bp0 = B + (size_t)col0 * ldb + khalf;
    bp1 = B + (size_t)col1 * ldb + khalf;
  } else {
    bp0 = B + (size_t)khalf * ldb + col0;
    bp1 = bp0 + 16;
  }

  v8f acc0 = {}, acc1 = {};
#pragma unroll 4
  for (int k0 = 0; k0 < K; k0 += 4) {
    v2f a = *(const v2f*)ap;  ap += 4;        // 8B aligned: lda even, khalf even
    v2f b0, b1;
    if (TRANSB) {
      b0 = *(const v2f*)bp0;  bp0 += 4;
      b1 = *(const v2f*)bp1;  bp1 += 4;
    } else {
      b0.x = bp0[0];  b0.y = bp0[ldb];
      b1.x = bp1[0];  b1.y = bp1[ldb];
      bp0 += (size_t)ldb * 4;
      bp1 += (size_t)ldb * 4;
    }
    acc0 = wmma_f32_4(a, b0, acc0);           // independent chains -> co-exec
    acc1 = wmma_f32_4(a, b1, acc1);
  }

  int mbase = tm * 16 + (lane >> 4) * 8;      // VGPR r: lanes<16 -> M=r, else M=r+8
#pragma unroll
  for (int r = 0; r < 8; ++r) {
    size_t o = (size_t)(mbase + r) * ldc;
    C[o + col0] = acc0[r];
    C[o + col1] = acc1[r];
  }
}

// ---------------------------------------------------------------------------
// 4) DFT basis tables: trigC[x,t]=cos(2*pi*x*t/T), trigS=sin.
// ---------------------------------------------------------------------------
__global__ void trig_kernel(float* __restrict__ trigC, float* __restrict__ trigS, int T) {
  int idx = blockIdx.x * blockDim.x + threadIdx.x;
  if (idx >= 32 * T) return;
  int x = idx / T, t = idx - x * T;
  float ang = 2.0f * PI_F * (float)x * (float)t / (float)T;
  float s, c;
  __sincosf(ang, &s, &c);
  trigC[idx] = c;
  trigS[idx] = s;
}

// ---------------------------------------------------------------------------
// 5) Fold irFFT-at-t*=95 phases into spectral weights:
//    Wc = g(cos*wr - sin*wi), Ws = g(cos*wi + sin*wr), g = (x?2:1)/T.
//    Written as a (16384 x 256) matrix: rows (x*256+i) -> Wc, +8192 -> Ws.
// ---------------------------------------------------------------------------
__global__ void build_w_kernel(const float* __restrict__ wr,
                               const float* __restrict__ wi,
                               float* __restrict__ Wm, int T) {
  int idx = blockIdx.x * blockDim.x + threadIdx.x;   // over 256*256*32
  if (idx >= 256 * 256 * 32) return;
  int x = idx & 31;
  int o = (idx >> 5) & 255;
  int i = idx >> 13;
  float r = wr[idx], im = wi[idx];
  float theta = 2.0f * PI_F * (float)x * 95.0f / (float)T;
  float g = ((x == 0) ? 1.0f : 2.0f) / (float)T;
  float s, c;
  __sincosf(theta, &s, &c);
  Wm[(size_t)(x * 256 + i) * 256 + o]        = g * (c * r - s * im);
  Wm[(size_t)(8192 + x * 256 + i) * 256 + o] = g * (c * im + s * r);
}

// ---------------------------------------------------------------------------
// 6) Causal trig correlations (ragged-K WMMA GEMM):
//    Cc[tau,x,i] = sum_{k<T-tau} trigC[x,k+tau]*P[k][i]  (Cs with sin).
//    One wave per (tau, 16x16 tile); cos/sin acc chains share the B fragment.
//    Scalar (SGPR) trip count via readfirstlane -> s_cmp loop, EXEC untouched.
//    Tail uses clamped in-bounds loads + cndmask zero-select (branch-free).
//    Output row tau of CcBuf: [Cc(x,i) | Cs(x,i)] length 16384.
// ---------------------------------------------------------------------------
__global__ __launch_bounds__(256) void cc_corr_kernel(
    const float* __restrict__ trigC, const float* __restrict__ trigS,
    const float* __restrict__ Pmat, float* __restrict__ CcBuf, int T) {
  int wave = uniform_i32(blockIdx.x * (blockDim.x >> 5) + (threadIdx.x >> 5));
  int total = T * 32;                         // T * (16 n-tiles) * (2 m-tiles)
  if (wave >= total) return;
  int tau = wave >> 5;                        // SGPR
  int rem = wave & 31;
  int tn = rem & 15;                          // i tile
  int tm = rem >> 4;                          // x tile (0/1)
  int lane = threadIdx.x & 31;
  int x = tm * 16 + (lane & 15);
  int i = tn * 16 + (lane & 15);
  int khalf = (lane >> 4) * 2;
  int Kt = T - tau;                           // SGPR
  int Kmain = Kt & ~3;                        // SGPR

  const float* tcrow = trigC + (size_t)x * T + tau;
  const float* tsrow = trigS + (size_t)x * T + tau;
  const float* tc = tcrow + khalf;
  const float* ts = tsrow + khalf;
  const float* pp = Pmat + (size_t)khalf * 256 + i;

  v8f accC = {}, accS = {};
#pragma unroll 4
  for (int k0 = 0; k0 < Kmain; k0 += 4) {
    v2f ac, as_, b;
    ac.x  = tc[0];  ac.y  = tc[1];  tc += 4;  // unaligned base -> scalar loads
    as_.x = ts[0];  as_.y = ts[1];  ts += 4;
    b.x   = pp[0];  b.y   = pp[256];          pp += 1024;
    accC = wmma_f32_4(ac, b, accC);
    accS = wmma_f32_4(as_, b, accS);
  }
  if (Kmain < Kt) {                           // <=1 tail iteration, branch-free
    int ka = Kmain + khalf;
    int k0c = min(ka,     Kt - 1);            // clamped, always in-bounds
    int k1c = min(ka + 1, Kt - 1);
    float m0 = (ka     < Kt) ? 1.f : 0.f;     // -> v_cndmask, EXEC untouched
    float m1 = (ka + 1 < Kt) ? 1.f : 0.f;
    v2f ac, as_, b;
    ac.x  = tcrow[k0c] * m0;  ac.y  = tcrow[k1c] * m1;
    as_.x = tsrow[k0c] * m0;  as_.y = tsrow[k1c] * m1;
    b.x   = Pmat[(size_t)k0c * 256 + i] * m0;
    b.y   = Pmat[(size_t)k1c * 256 + i] * m1;
    accC = wmma_f32_4(ac, b, accC);
    accS = wmma_f32_4(as_, b, accS);
  }

  int xbase = tm * 16 + (lane >> 4) * 8;
  size_t rowo = (size_t)tau * 16384;
#pragma unroll
  for (int r = 0; r < 8; ++r) {
    CcBuf[rowo + (xbase + r) * 256 + i]        = accC[r];
    CcBuf[rowo + 8192 + (xbase + r) * 256 + i] = accS[r];
  }
}

// ---------------------------------------------------------------------------
// 7) Assemble the combined 96x720 operator with MLP scale mix.
// ---------------------------------------------------------------------------
__global__ void combine_kernel(const float* __restrict__ Gs,
                               const float* __restrict__ mlp_w,
                               float* __restrict__ Gall) {
  int idx = blockIdx.x * blockDim.x + threadIdx.x;   // 96*720
  if (idx >= 96 * 720) return;
  int t = idx / 720, p = idx - t * 720;
  const int Toff[3] = {0, 96 * 96, 96 * 96 + 96 * 192};
  const int Ts[3] = {96, 192, 384};
  float acc = 0.f;
  for (int s = 0; s < 3; ++s) {
    int start = 720 - Ts[s];
    if (p >= start)
      acc += mlp_w[s] * Gs[Toff[s] + t * Ts[s] + (p - start)];
  }
  Gall[idx] = acc;
}

// ---------------------------------------------------------------------------
// 8) Apply operator + denormalize: out[b,t,e].
// ---------------------------------------------------------------------------
__global__ void apply_kernel(const float* __restrict__ Gall,
                             const float* __restrict__ xn,
                             const float* __restrict__ meanA,
                             const float* __restrict__ stdA,
                             const float* __restrict__ mlp_b,
                             float* __restrict__ out) {
  int idx = blockIdx.x * blockDim.x + threadIdx.x;   // 16*96*64
  if (idx >= 16 * 96 * 64) return;
  int e = idx & 63;
  int t = (idx >> 6) % 96;
  int b = idx / (96 * 64);
  const float* xp = xn + (size_t)b * 720 * 64 + e;
  const float* g = Gall + (size_t)t * 720;
  float s = 0.f;
#pragma unroll 4
  for (int p = 0; p < 720; ++p) s += g[p] * xp[(size_t)p * 64];
  int ch = b * 64 + e;
  out[idx] = (s + mlp_b[0]) * stdA[ch] + meanA[ch];
}

// ---------------------------------------------------------------------------
static void launch_gemm(const float* A, int lda, const float* B, int ldb,
                        float* C, int ldc, int M, int N, int K,
                        int transB, hipStream_t st) {
  int tiles = (M / 16) * (N / 32);
  int blocks = (tiles + 7) / 8;
  if (transB)
    gemm_wmma_f32<1><<<dim3(blocks), dim3(256), 0, st>>>(A, lda, B, ldb, C, ldc, N, K, tiles);
  else
    gemm_wmma_f32<0><<<dim3(blocks), dim3(256), 0, st>>>(A, lda, B, ldb, C, ldc, N, K, tiles);
}

extern "C" void kernel_launch(void* const* d_in, const int* in_sizes, int n_in,
                              void* d_out, int out_size, void* d_ws, size_t ws_size,
                              hipStream_t stream) {
  const float* x     = (const float*)d_in[0];   // (16,720,64)
  const float* wr    = (const float*)d_in[1];   // (3,256,256,32)
  const float* wi    = (const float*)d_in[2];   // (3,256,256,32)
  const float* mlp_w = (const float*)d_in[3];   // (1,3)
  const float* mlp_b = (const float*)d_in[4];   // (1,)
  const float* A     = (const float*)d_in[5];   // (3,256,256)
  const float* Bv    = (const float*)d_in[6];   // (3,256)
  const float* ev    = (const float*)d_in[7];   // (3,96,256)
  float* out = (float*)d_out;
  float* ws = (float*)d_ws;

  // workspace carve-up (floats) ~ 47 MB total
  size_t off = 0;
  auto alloc = [&](size_t n) { float* p = ws + off; off += n; return p; };
  float* xn    = alloc((size_t)16 * 720 * 64);
  float* meanA = alloc(1024);
  float* stdA  = alloc(1024);
  float* Pmat  = alloc((size_t)384 * 256);        // reused per scale (padded)
  float* Qt0   = alloc(65536);
  float* Qt1   = alloc(65536);
  float* trigC = alloc((size_t)32 * 384);
  float* trigS = alloc((size_t)32 * 384);
  float* Wm    = alloc((size_t)16384 * 256);      // reused per scale
  float* CcBuf = alloc((size_t)384 * 16384);      // reused per scale
  float* M2    = alloc((size_t)384 * 256);        // reused per scale
  float* Gs    = alloc((size_t)96 * (96 + 192 + 384));
  float* Gall  = alloc((size_t)96 * 720);
  (void)in_sizes; (void)n_in; (void)out_size; (void)ws_size;

  // 1) normalization (1024 waves)
  hipLaunchKernelGGL(norm_kernel, dim3(128), dim3(256), 0, stream, x, xn, meanA, stdA);

  const int GsOff[3] = {0, 96 * 96, 96 * 96 + 96 * 192};

  for (int s = 0; s < 3; ++s) {
    const int T = 96 << s;                       // 96, 192, 384
    const int padRows = ((T + 63) / 64) * 64;    // 128, 192, 384
    const float* Ad = A + (size_t)s * 65536;
    const float* Bd = Bv + (size_t)s * 256;

    // trig tables
    hipLaunchKernelGGL(trig_kernel, dim3((32 * T + 255) / 256), dim3(256), 0, stream,
                       trigC, trigS, T);

    // P[0..63] sequentially, Ad in 256KB dynamic LDS
    hipLaunchKernelGGL(seqP_kernel, dim3(1), dim3(256),
                       (65536 + 256) * sizeof(float), stream, Ad, Bd, Pmat, 64);

    // Ad^64 via 6 squarings (ping-pong): t0=Ad^2, t1=Ad^4, ... t1=Ad^64
    launch_gemm(Ad, 256, Ad, 256, Qt0, 256, 256, 256, 256, 0, stream);
    launch_gemm(Qt0, 256, Qt0, 256, Qt1, 256, 256, 256, 256, 0, stream);
    launch_gemm(Qt1, 256, Qt1, 256, Qt0, 256, 256, 256, 256, 0, stream);
    launch_gemm(Qt0, 256, Qt0, 256, Qt1, 256, 256, 256, 256, 0, stream);
    launch_gemm(Qt1, 256, Qt1, 256, Qt0, 256, 256, 256, 256, 0, stream);
    launch_gemm(Qt0, 256, Qt0, 256, Qt1, 256, 256, 256, 256, 0, stream);

    // advance P in 64-row blocks: block_j = block_{j-1} @ (Ad^64)^T
    for (int j = 1; j < padRows / 64; ++j)
      launch_gemm(Pmat + (size_t)(j - 1) * 64 * 256, 256, Qt1, 256,
                  Pmat + (size_t)j * 64 * 256, 256, 64, 256, 256, 1, stream);

    // folded spectral weights (16384 x 256)
    hipLaunchKernelGGL(build_w_kernel, dim3((256 * 256 * 32 + 255) / 256), dim3(256),
                       0, stream, wr + (size_t)s * 2097152, wi + (size_t)s * 2097152, Wm, T);

    // causal correlations -> CcBuf (T x 16384), T*32 waves
    hipLaunchKernelGGL(cc_corr_kernel, dim3(T * 4), dim3(256), 0, stream,
                       trigC, trigS, Pmat, CcBuf, T);

    // big contraction: M2 (T x 256) = CcBuf @ Wm  (K=16384)
    launch_gemm(CcBuf, 16384, Wm, 256, M2, 256, T, 256, 16384, 0, stream);

    // Legendre decode: Gs_s (96 x T) = ev_s (96x256) @ M2^T
    launch_gemm(ev + (size_t)s * 96 * 256, 256, M2, 256,
                Gs + GsOff[s], T, 96, T, 256, 1, stream);
  }

  // combined operator + MLP mix, then apply + denormalize
  hipLaunchKernelGGL(combine_kernel, dim3((96 * 720 + 255) / 256), dim3(256), 0, stream,
                     Gs, mlp_w, Gall);
  hipLaunchKernelGGL(apply_kernel, dim3((16 * 96 * 64 + 255) / 256), dim3(256), 0, stream,
                     Gall, xn, meanA, stdA, mlp_b, out);
}